// Block1_3642132267801
// MI455X (gfx1250) — compile-verified
//
#include <hip/hip_runtime.h>

// ---------------------------------------------------------------------------
// MI455X (gfx1250) edge-filter pipeline.
//
// Roofline: 32MB pan read + 32MB pan_lp write/read (L2-resident) + 192MB out
// -> ~290MB => ~12.5us @ 23.3TB/s. Pass 2 evaluates the 7-filter x 9-tap bank
// for TWO 16-pixel strips per iteration with 12 chained V_WMMA_F32_16X16X4_F32
// into a single 16x16 f32 accumulator: strip-1 filters occupy A rows 0-6
// (D lands in lanes 0-15), strip-2 filters occupy A rows 8-14 (D lands in
// lanes 16-31). Epilogue runs on all 32 lanes with no EXEC divergence.
// ---------------------------------------------------------------------------

typedef __attribute__((ext_vector_type(2))) float v2f;
typedef __attribute__((ext_vector_type(8))) float v8f;

#define IMG_H 2048
#define IMG_W 2048
#define IMG_HW (IMG_H * IMG_W)

// Filter bank, row-major 3x3 flattened to taps t = 3*(dy+1)+(dx+1), padded to K=12.
// 0=SOBEL_X 1=SOBEL_Y (raw taps), 2=PREWITT_X 3=PREWITT_Y 4=ROBERTS_X
// 5=ROBERTS_Y 6=LAPLACE (floored taps).
__constant__ float c_wt[7][12] = {
    {-1.f, -2.f, -1.f,  0.f,  0.f,  0.f,  1.f,  2.f,  1.f, 0.f, 0.f, 0.f}, // SOBEL_X
    {-1.f,  0.f,  1.f, -2.f,  0.f,  2.f, -1.f,  0.f,  1.f, 0.f, 0.f, 0.f}, // SOBEL_Y
    { 1.f,  1.f,  1.f,  0.f,  0.f,  0.f, -1.f, -1.f, -1.f, 0.f, 0.f, 0.f}, // PREWITT_X
    {-1.f,  0.f,  1.f, -1.f,  0.f,  1.f, -1.f,  0.f,  1.f, 0.f, 0.f, 0.f}, // PREWITT_Y
    {-1.f,  0.f,  0.f,  0.f,  1.f,  0.f,  0.f,  0.f,  0.f, 0.f, 0.f, 0.f}, // ROBERTS_X
    { 0.f, -1.f,  0.f,  1.f,  0.f,  0.f,  0.f,  0.f,  0.f, 0.f, 0.f, 0.f}, // ROBERTS_Y
    { 2.f,  0.f,  2.f,  0.f, -8.f,  0.f,  2.f,  0.f,  2.f, 0.f, 0.f, 0.f}, // LAPLACE
};

#define WMMA_F32_K4(Av, Bv, Cv)                                              \
    __builtin_amdgcn_wmma_f32_16x16x4_f32(false, (Av), false, (Bv),          \
                                          (short)0, (Cv), false, false)

// ---------------------------------------------------------------------------
// Pass 1: gauss3x3(reflect) -> ws_lp ; per-batch max (atomicMax on float
// bits; all values >= 0) ; diff_y / diff_x output channels 0,1.
// ---------------------------------------------------------------------------
__global__ void pass1_gauss_diff_max(const float* __restrict__ pan,
                                     float* __restrict__ out,
                                     float* __restrict__ ws_lp,
                                     unsigned int* __restrict__ ws_max) {
    const int b   = blockIdx.y;
    const int idx = blockIdx.x * blockDim.x + threadIdx.x;   // HW divisible by 256
    const int y   = idx / IMG_W;
    const int x   = idx - y * IMG_W;

    const int ym1 = (y == 0) ? 1 : y - 1;
    const int yp1 = (y == IMG_H - 1) ? IMG_H - 2 : y + 1;
    const int xm1 = (x == 0) ? 1 : x - 1;
    const int xp1 = (x == IMG_W - 1) ? IMG_W - 2 : x + 1;

    const float* pb = pan + (size_t)b * IMG_HW;
    const float a00 = pb[(size_t)ym1 * IMG_W + xm1];
    const float a01 = pb[(size_t)ym1 * IMG_W + x];
    const float a02 = pb[(size_t)ym1 * IMG_W + xp1];
    const float a10 = pb[(size_t)y   * IMG_W + xm1];
    const float a11 = pb[(size_t)y   * IMG_W + x];
    const float a12 = pb[(size_t)y   * IMG_W + xp1];
    const float a20 = pb[(size_t)yp1 * IMG_W + xm1];
    const float a21 = pb[(size_t)yp1 * IMG_W + x];
    const float a22 = pb[(size_t)yp1 * IMG_W + xp1];

    const float g = 0.0625f * (a00 + a02 + a20 + a22)
                  + 0.125f  * (a01 + a10 + a12 + a21)
                  + 0.25f   * a11;

    ws_lp[(size_t)b * IMG_HW + idx] = g;

    // Forward differences, replicate pad on left/top. For y>0 / x>0 the
    // reflect-index loads above equal the y-1 / x-1 taps.
    const float dy = (y > 0) ? (a11 - a01) : 0.0f;
    const float dx = (x > 0) ? (a11 - a10) : 0.0f;

    float* outb = out + (size_t)b * 6 * IMG_HW;
    outb[(size_t)0 * IMG_HW + idx] = dy;
    outb[(size_t)1 * IMG_HW + idx] = dx;

    __shared__ float sm[256];
    const int tid = threadIdx.x;
    sm[tid] = g;
    __syncthreads();
    for (int s = 128; s > 0; s >>= 1) {
        if (tid < s) sm[tid] = fmaxf(sm[tid], sm[tid + s]);
        __syncthreads();
    }
    if (tid == 0) atomicMax(ws_max + b, __float_as_uint(sm[0]));
}

// ---------------------------------------------------------------------------
// Pass 2. Per wave: one image row, 64 iterations of 32 pixels.
// A layout (ISA 7.12.2, 32-bit A 16x4): lane holds row M=lane&15; lanes 0-15
// carry K={4b,4b+1}, lanes 16-31 K={4b+2,4b+3}. B mirrors with N=lane&15.
// D layout: VGPR f = row f (lanes 0-15) and row f+8 (lanes 16-31).
// ---------------------------------------------------------------------------
__global__ void pass2_edges_wmma(const float* __restrict__ ws_lp,
                                 const unsigned int* __restrict__ ws_max,
                                 float* __restrict__ out) {
    const int b    = blockIdx.y;
    const int lane = threadIdx.x & 31;
    const int wave = threadIdx.x >> 5;
    const int y    = blockIdx.x * (blockDim.x >> 5) + wave;   // one row per wave
    const bool hi  = (lane >= 16);
    const int  n   = lane & 15;

    const float norm  = __uint_as_float(ws_max[b]);
    const float scale = norm / 255.0f;
    const float inv   = 255.0f / norm;       // hoisted: 1 mul/tap, no per-tap divide

    // Loop-invariant A slices. Strip-1 bank in rows 0-6; strip-2 bank in rows 8-14.
    v2f a_s1[3], a_f1[3], a_s2[3], a_f2[3];
    {
        const int m = n;                       // A row index for this lane
        for (int blk = 0; blk < 3; ++blk) {
            const int k = blk * 4 + (hi ? 2 : 0);
            v2f z; z.x = 0.f; z.y = 0.f;
            a_s1[blk] = z; a_f1[blk] = z; a_s2[blk] = z; a_f2[blk] = z;
            if (m < 2)             { a_s1[blk].x = c_wt[m][k];     a_s1[blk].y = c_wt[m][k + 1]; }
            if (m >= 2 && m < 7)   { a_f1[blk].x = c_wt[m][k];     a_f1[blk].y = c_wt[m][k + 1]; }
            if (m >= 8 && m < 10)  { a_s2[blk].x = c_wt[m - 8][k]; a_s2[blk].y = c_wt[m - 8][k + 1]; }
            if (m >= 10 && m < 15) { a_f2[blk].x = c_wt[m - 8][k]; a_f2[blk].y = c_wt[m - 8][k + 1]; }
        }
    }

    const int ym1 = (y == 0) ? 1 : y - 1;
    const int yp1 = (y == IMG_H - 1) ? IMG_H - 2 : y + 1;
    const float* lp   = ws_lp + (size_t)b * IMG_HW;
    const float* rowm = lp + (size_t)ym1 * IMG_W;
    const float* row0 = lp + (size_t)y   * IMG_W;
    const float* rowp = lp + (size_t)yp1 * IMG_W;

    float* outb  = out + (size_t)b * 6 * IMG_HW;
    float* orow2 = outb + (size_t)2 * IMG_HW + (size_t)y * IMG_W;  // roberts
    float* orow3 = outb + (size_t)3 * IMG_HW + (size_t)y * IMG_W;  // prewitt
    float* orow4 = outb + (size_t)4 * IMG_HW + (size_t)y * IMG_W;  // sobel
    float* orow5 = outb + (size_t)5 * IMG_HW + (size_t)y * IMG_W;  // laplacian

    for (int xs = 0; xs < IMG_W; xs += 32) {
        // Strip 1: x1 in [xs, xs+16); strip 2: x2 in [xs+16, xs+32).
        const int x1   = xs + n;
        const int x2   = x1 + 16;
        const int x1m1 = (x1 == 0) ? 1 : x1 - 1;          // left edge only in strip 1
        const int x1p1 = x1 + 1;                          // x1 <= W-17, never wraps
        const int x2m1 = x2 - 1;                          // x2 >= 16, never wraps
        const int x2p1 = (x2 == IMG_W - 1) ? IMG_W - 2 : x2 + 1;  // right edge only in strip 2

        if (xs + 64 < IMG_W) {                 // global_prefetch_b8, 2 iterations (4 strips) ahead
            __builtin_prefetch(row0 + xs + 64, 0, 1);
            __builtin_prefetch(rowm + xs + 64, 0, 1);
            __builtin_prefetch(rowp + xs + 64, 0, 1);
        }

        // Tap split: lanes 0-15 supply K taps {0,1,4,5,8}, lanes 16-31 {2,3,6,7}
        // for BOTH strips. Address selects keep EXEC full.
        const float A0 = rowm[hi ? x1p1 : x1m1];               // t2 : t0
        const float A1 = (hi ? row0 : rowm)[hi ? x1m1 : x1];   // t3 : t1
        const float A2 = (hi ? rowp : row0)[hi ? x1m1 : x1];   // t6 : t4
        const float A3 = (hi ? rowp : row0)[hi ? x1 : x1p1];   // t7 : t5
        const float A4 = rowp[x1p1];                           // t8 (lo lanes)

        const float B0 = rowm[hi ? x2p1 : x2m1];
        const float B1 = (hi ? row0 : rowm)[hi ? x2m1 : x2];
        const float B2 = (hi ? rowp : row0)[hi ? x2m1 : x2];
        const float B3 = (hi ? rowp : row0)[hi ? x2 : x2p1];
        const float B4 = rowp[x2p1];

        const float p0 = A0 * inv, p1 = A1 * inv, p2 = A2 * inv, p3 = A3 * inv;
        const float p4 = hi ? 0.0f : A4 * inv;
        const float q0 = B0 * inv, q1 = B1 * inv, q2 = B2 * inv, q3 = B3 * inv;
        const float q4 = hi ? 0.0f : B4 * inv;

        v2f P0, P1, P2, Pf0, Pf1, Pf2, Q0, Q1, Q2, Qf0, Qf1, Qf2;
        P0.x = p0;          P0.y = p1;
        P1.x = p2;          P1.y = p3;
        P2.x = p4;          P2.y = 0.0f;
        Pf0.x = floorf(p0); Pf0.y = floorf(p1);
        Pf1.x = floorf(p2); Pf1.y = floorf(p3);
        Pf2.x = floorf(p4); Pf2.y = 0.0f;
        Q0.x = q0;          Q0.y = q1;
        Q1.x = q2;          Q1.y = q3;
        Q2.x = q4;          Q2.y = 0.0f;
        Qf0.x = floorf(q0); Qf0.y = floorf(q1);
        Qf1.x = floorf(q2); Qf1.y = floorf(q3);
        Qf2.x = floorf(q4); Qf2.y = 0.0f;

        v8f acc = {0.f, 0.f, 0.f, 0.f, 0.f, 0.f, 0.f, 0.f};
        // Strip 1 -> rows 0-6 (lanes 0-15 of D)
        acc = WMMA_F32_K4(a_s1[0], P0,  acc);
        acc = WMMA_F32_K4(a_s1[1], P1,  acc);
        acc = WMMA_F32_K4(a_s1[2], P2,  acc);
        acc = WMMA_F32_K4(a_f1[0], Pf0, acc);
        acc = WMMA_F32_K4(a_f1[1], Pf1, acc);
        acc = WMMA_F32_K4(a_f1[2], Pf2, acc);
        // Strip 2 -> rows 8-14 (lanes 16-31 of D)
        acc = WMMA_F32_K4(a_s2[0], Q0,  acc);
        acc = WMMA_F32_K4(a_s2[1], Q1,  acc);
        acc = WMMA_F32_K4(a_s2[2], Q2,  acc);
        acc = WMMA_F32_K4(a_f2[0], Qf0, acc);
        acc = WMMA_F32_K4(a_f2[1], Qf1, acc);
        acc = WMMA_F32_K4(a_f2[2], Qf2, acc);

        // Epilogue on all 32 lanes: lane owns pixel xs + n (+16 for hi half).
        const float e_sx = rintf(fminf(fabsf(acc[0]), 255.0f));
        const float e_sy = rintf(fminf(fabsf(acc[1]), 255.0f));
        const float e_px = rintf(fminf(fabsf(acc[2]), 255.0f));
        const float e_py = rintf(fminf(fabsf(acc[3]), 255.0f));
        const float e_rx = rintf(fminf(fabsf(acc[4]), 255.0f));
        const float e_ry = rintf(fminf(fabsf(acc[5]), 255.0f));
        const float e_lp = rintf(fminf(fabsf(acc[6]), 255.0f));

        const float sobel   = rintf(0.5f * e_sx + 0.5f * e_sy) * scale;
        const float prewitt = rintf(0.5f * e_px + 0.5f * e_py) * scale;
        const float roberts = rintf(0.5f * e_rx + 0.5f * e_ry) * scale;
        const float laplace = e_lp * scale;

        const int xout = hi ? x2 : x1;
        orow2[xout] = roberts;
        orow3[xout] = prewitt;
        orow4[xout] = sobel;
        orow5[xout] = laplace;
    }
}

// ---------------------------------------------------------------------------
extern "C" void kernel_launch(void* const* d_in, const int* in_sizes, int n_in,
                              void* d_out, int out_size, void* d_ws, size_t ws_size,
                              hipStream_t stream) {
    const float* pan = (const float*)d_in[0];
    float* out = (float*)d_out;
    const int B = in_sizes[0] / IMG_HW;   // = 2

    // ws layout: [0,256): per-batch max slots (uint float-bits); [256,...): pan_lp.
    unsigned int* ws_max = (unsigned int*)d_ws;
    float* ws_lp = (float*)((char*)d_ws + 256);

    hipMemsetAsync(ws_max, 0, (size_t)B * sizeof(unsigned int), stream);

    dim3 g1(IMG_HW / 256, B);
    pass1_gauss_diff_max<<<g1, 256, 0, stream>>>(pan, out, ws_lp, ws_max);

    dim3 g2(IMG_H / 4, B);                 // 128 threads = 4 waves = 4 rows/block
    pass2_edges_wmma<<<g2, 128, 0, stream>>>(ws_lp, ws_max, out);
}